// OriginLossfuc_64398739636467
// MI455X (gfx1250) — compile-verified
//
#include <hip/hip_runtime.h>
#include <hip/hip_bf16.h>

// Triplet cosine loss, diagonal-only formulation.
// Stage 1: one wave32 per row streams a/p/n rows (b128 loads), accumulates
//          5 dot products in f32, butterfly-reduces across 32 lanes, folds
//          8 rows/block into one partial -> d_ws[block].
// Stage 2: one block deterministically reduces the 1024 partials; the wave
//          level 64-way reduction is done with V_WMMA_F32_16X16X4_F32
//          (A = ones => layout-independent sum of all B slots).

typedef float v2f __attribute__((ext_vector_type(2)));
typedef float v8f __attribute__((ext_vector_type(8)));

#define MARGIN_F 0.1f
#define D_DIM 1024
#define ROWS_PER_BLOCK 8   // 8 waves * 1 row/wave

__global__ __launch_bounds__(256) void triplet_rows_kernel(
    const float* __restrict__ a, const float* __restrict__ p,
    const float* __restrict__ n, float* __restrict__ partial)
{
    const int lane = threadIdx.x & 31;
    const int wave = threadIdx.x >> 5;
    const int row  = blockIdx.x * ROWS_PER_BLOCK + wave;

    const float4* a4 = (const float4*)(a + (size_t)row * D_DIM);
    const float4* p4 = (const float4*)(p + (size_t)row * D_DIM);
    const float4* n4 = (const float4*)(n + (size_t)row * D_DIM);

    float s_ap = 0.f, s_an = 0.f, s_aa = 0.f, s_pp = 0.f, s_nn = 0.f;

    // 1024 floats/row: lane handles float4 indices lane + 32*j, j=0..7.
    // Each wave-iteration touches 3 * 512 contiguous bytes per array.
#pragma unroll
    for (int j = 0; j < D_DIM / 128; ++j) {
        const int idx = lane + 32 * j;
        float4 va = a4[idx];
        float4 vp = p4[idx];
        float4 vn = n4[idx];

        s_ap = fmaf(va.x, vp.x, s_ap); s_ap = fmaf(va.y, vp.y, s_ap);
        s_ap = fmaf(va.z, vp.z, s_ap); s_ap = fmaf(va.w, vp.w, s_ap);

        s_an = fmaf(va.x, vn.x, s_an); s_an = fmaf(va.y, vn.y, s_an);
        s_an = fmaf(va.z, vn.z, s_an); s_an = fmaf(va.w, vn.w, s_an);

        s_aa = fmaf(va.x, va.x, s_aa); s_aa = fmaf(va.y, va.y, s_aa);
        s_aa = fmaf(va.z, va.z, s_aa); s_aa = fmaf(va.w, va.w, s_aa);

        s_pp = fmaf(vp.x, vp.x, s_pp); s_pp = fmaf(vp.y, vp.y, s_pp);
        s_pp = fmaf(vp.z, vp.z, s_pp); s_pp = fmaf(vp.w, vp.w, s_pp);

        s_nn = fmaf(vn.x, vn.x, s_nn); s_nn = fmaf(vn.y, vn.y, s_nn);
        s_nn = fmaf(vn.z, vn.z, s_nn); s_nn = fmaf(vn.w, vn.w, s_nn);
    }

    // wave32 butterfly reduction (deterministic, fixed order)
#pragma unroll
    for (int off = 16; off > 0; off >>= 1) {
        s_ap += __shfl_xor(s_ap, off, 32);
        s_an += __shfl_xor(s_an, off, 32);
        s_aa += __shfl_xor(s_aa, off, 32);
        s_pp += __shfl_xor(s_pp, off, 32);
        s_nn += __shfl_xor(s_nn, off, 32);
    }

    __shared__ float sh[ROWS_PER_BLOCK];
    if (lane == 0) {
        float ap = s_ap / (sqrtf(s_aa) * sqrtf(s_pp));
        float an = s_an / (sqrtf(s_aa) * sqrtf(s_nn));
        sh[wave] = fmaxf(0.0f, an - ap + MARGIN_F);
    }
    __syncthreads();
    if (threadIdx.x == 0) {
        float t = 0.f;
#pragma unroll
        for (int i = 0; i < ROWS_PER_BLOCK; ++i) t += sh[i];
        partial[blockIdx.x] = t;
    }
}

__global__ __launch_bounds__(128) void final_reduce_kernel(
    const float* __restrict__ partial, float* __restrict__ out,
    int nparts, float inv_n)
{
    const int t = threadIdx.x;

    // Fixed-order serial accumulation: deterministic across replays.
    float acc0 = 0.f, acc1 = 0.f;
    for (int i = t; i < nparts; i += 256)       acc0 += partial[i];
    for (int i = t + 128; i < nparts; i += 256) acc1 += partial[i];

    // Wave-level 64-element reduction via WMMA (f32 path, exact precision).
    // A = all ones: every slot of the 16x4 A matrix is 1.0 regardless of
    // VGPR layout, so D[m][n] = colsum(B)[n]. Summing D VGPR0 over the 32
    // lanes counts every B slot exactly twice (rows M=0 and M=8 both carry
    // the 16 column sums) => wave_total = 0.5 * lane_sum(D[0]).
    v2f av; av.x = 1.0f; av.y = 1.0f;
    v2f bv; bv.x = acc0; bv.y = acc1;
    v8f cv = {};
    v8f dv = __builtin_amdgcn_wmma_f32_16x16x4_f32(
        /*neg_a=*/false, av, /*neg_b=*/false, bv,
        /*c_mod=*/(short)0, cv, /*reuse_a=*/false, /*reuse_b=*/false);

    float v = dv[0];
#pragma unroll
    for (int off = 16; off > 0; off >>= 1) v += __shfl_xor(v, off, 32);
    v *= 0.5f;  // wave total

    __shared__ float sh[4];
    if ((t & 31) == 0) sh[t >> 5] = v;
    __syncthreads();
    if (t == 0) out[0] = (sh[0] + sh[1] + sh[2] + sh[3]) * inv_n;
}

extern "C" void kernel_launch(void* const* d_in, const int* in_sizes, int n_in,
                              void* d_out, int out_size, void* d_ws, size_t ws_size,
                              hipStream_t stream) {
    const float* a = (const float*)d_in[0];
    const float* p = (const float*)d_in[1];
    const float* n = (const float*)d_in[2];
    float* out = (float*)d_out;
    float* ws  = (float*)d_ws;

    const int nrows   = in_sizes[0] / D_DIM;        // 8192
    const int nblocks = nrows / ROWS_PER_BLOCK;     // 1024

    triplet_rows_kernel<<<nblocks, 256, 0, stream>>>(a, p, n, ws);
    final_reduce_kernel<<<1, 128, 0, stream>>>(ws, out, nblocks,
                                               1.0f / (float)nrows);
}